// CroquetGNN_43215960932691
// MI455X (gfx1250) — compile-verified
//
#include <hip/hip_runtime.h>

typedef __attribute__((ext_vector_type(2))) float v2f;
typedef __attribute__((ext_vector_type(8))) float v8f;

__device__ __forceinline__ void fadd_atomic(float* p, float v) {
    // hardware global_atomic_add_f32 (no-return form; result unused)
    unsafeAtomicAdd(p, v);
}

// ---- degree / normalization -------------------------------------------------

__global__ void k_deg_init(float* __restrict__ deg, int n) {
    int i = blockIdx.x * blockDim.x + threadIdx.x;
    if (i < n) deg[i] = 1.0f;                    // self-loop contribution
}

__global__ void k_deg_edges(const int* __restrict__ dst, float* __restrict__ deg, int E) {
    int e = blockIdx.x * blockDim.x + threadIdx.x;
    if (e < E) fadd_atomic(&deg[dst[e]], 1.0f);
}

__global__ void k_rsqrt_inplace(float* __restrict__ d, int n) {
    int i = blockIdx.x * blockDim.x + threadIdx.x;
    if (i < n) d[i] = rsqrtf(d[i]);              // deg >= 1 always (self loops)
}

// ---- layer 1 linear: h1 = x @ W1, hs1 = h1*dis, acc1 seeded with self-loop --
// One wave per 16-row tile; 16x4 (f32) x 4x16 (f32) -> 16x16 via
// V_WMMA_F32_16X16X4_F32.

__global__ void k_lin1_wmma(const float* __restrict__ x, const float* __restrict__ W1,
                            const float* __restrict__ dis,
                            float* __restrict__ hs1, float* __restrict__ acc1,
                            int ntile, int n) {
    const int lane = threadIdx.x & 31;
    const int wave = threadIdx.x >> 5;
    const int tile = blockIdx.x * (blockDim.x >> 5) + wave;
    if (tile >= ntile) return;

    const int m  = lane & 15;        // row (A) / col (B,D) within tile
    const int hi = lane >> 4;        // lane half selects K pair {0,1} vs {2,3}
    const int kb = hi * 2;
    const int base = tile * 16;

    int g  = base + m;
    int gc = g < n ? g : n - 1;      // clamp tail loads (EXEC must stay full)
    float dloc = dis[gc];            // lane m holds dis[base+m] (halves duplicate)

    v2f a, b;
    a.x = x[gc * 4 + kb];
    a.y = x[gc * 4 + kb + 1];
    b.x = W1[kb * 16 + m];           // W1 is [4,16] row-major: B[k][n]
    b.y = W1[(kb + 1) * 16 + m];

    v8f c = {};
    c = __builtin_amdgcn_wmma_f32_16x16x4_f32(false, a, false, b, (short)0, c, false, false);

    // D layout: VGPR r holds row M = r + 8*hi, col = lane&15
    if (base + 16 <= n) {            // fast path: full tile, branch-free
#pragma unroll
        for (int r = 0; r < 8; ++r) {
            int row = base + r + 8 * hi;
            float dr = __shfl(dloc, r + 8 * hi);   // dis[row] via lane broadcast
            float hv = c[r] * dr;                  // hs = h * dis[src-side]
            hs1[row * 16 + m]  = hv;
            acc1[row * 16 + m] = hv;               // self-loop seed: +hs[i]
        }
    } else {                         // tail tile only
#pragma unroll
        for (int r = 0; r < 8; ++r) {
            int row = base + r + 8 * hi;
            float dr = __shfl(dloc, r + 8 * hi);
            if (row < n) {
                float hv = c[r] * dr;
                hs1[row * 16 + m]  = hv;
                acc1[row * 16 + m] = hv;
            }
        }
    }
}

// ---- layer-1 edge scatter: acc1[dst] += hs1[src], 4 lanes per edge ----------

__global__ void k_edge_scatter16(const int* __restrict__ src, const int* __restrict__ dst,
                                 const float* __restrict__ hs1, float* __restrict__ acc1,
                                 int E) {
    int tid = blockIdx.x * blockDim.x + threadIdx.x;   // E*4 = 25.6M fits int
    if (tid >= E * 4) return;
    int e = tid >> 2;
    int q = (tid & 3) * 4;
    int s = src[e], d = dst[e];                        // quad-broadcast L1 hits
    const float4 v = *(const float4*)(hs1 + s * 16 + q);   // 16B aligned, L2-resident
    float* ap = acc1 + d * 16 + q;
    fadd_atomic(ap + 0, v.x);
    fadd_atomic(ap + 1, v.y);
    fadd_atomic(ap + 2, v.z);
    fadd_atomic(ap + 3, v.w);
}

// ---- layer 2 linear fused with relu/bias/final-scale of layer 1 -------------
// h1 = relu(dis*acc1 + b1) built directly into WMMA A operands; K=16 done as
// four chained 16x16x4 f32 WMMAs; W2 zero-padded 16x2 -> 16x16 (cols >= 2 are 0).

__global__ void k_lin2_wmma(const float* __restrict__ acc1, const float* __restrict__ b1,
                            const float* __restrict__ W2, const float* __restrict__ dis,
                            float* __restrict__ hs2, float* __restrict__ acc2,
                            int ntile, int n) {
    const int lane = threadIdx.x & 31;
    const int wave = threadIdx.x >> 5;
    const int tile = blockIdx.x * (blockDim.x >> 5) + wave;
    if (tile >= ntile) return;

    const int m  = lane & 15;
    const int hi = lane >> 4;
    const int base = tile * 16;

    int g  = base + m;
    int gc = g < n ? g : n - 1;
    float dg = dis[gc];

    v8f c = {};
#pragma unroll
    for (int ch = 0; ch < 4; ++ch) {
        const int kb = ch * 4 + hi * 2;
        v2f a, b;
        a.x = fmaxf(dg * acc1[gc * 16 + kb]     + b1[kb],     0.0f);
        a.y = fmaxf(dg * acc1[gc * 16 + kb + 1] + b1[kb + 1], 0.0f);
        b.x = (m < 2) ? W2[kb * 2 + m]       : 0.0f;   // W2 is [16,2] row-major
        b.y = (m < 2) ? W2[(kb + 1) * 2 + m] : 0.0f;
        c = __builtin_amdgcn_wmma_f32_16x16x4_f32(false, a, false, b, (short)0, c, false, false);
    }

    // only cols 0,1 of D are meaningful; lanes with m<2 write
    bool full = (base + 16 <= n);
    if (m < 2) {
#pragma unroll
        for (int r = 0; r < 8; ++r) {
            int row = base + r + 8 * hi;
            float dr = __shfl(dg, r + 8 * hi);
            if (full || row < n) {
                float hv = c[r] * dr;
                hs2[row * 2 + m]  = hv;
                acc2[row * 2 + m] = hv;              // self-loop seed
            }
        }
    }
}

// ---- layer-2 edge scatter: acc2[dst] += hs2[src], 1 thread per edge ---------

__global__ void k_edge_scatter2(const int* __restrict__ src, const int* __restrict__ dst,
                                const float* __restrict__ hs2, float* __restrict__ acc2,
                                int E) {
    int e = blockIdx.x * blockDim.x + threadIdx.x;
    if (e >= E) return;
    int s = src[e], d = dst[e];
    const float2 v = *(const float2*)(hs2 + s * 2);    // b64 gather, L2-resident
    float* ap = acc2 + d * 2;
    fadd_atomic(ap + 0, v.x);
    fadd_atomic(ap + 1, v.y);
}

// ---- final: out = dis * acc2 + b2 -------------------------------------------

__global__ void k_finish(const float* __restrict__ acc2, const float* __restrict__ dis,
                         const float* __restrict__ b2, float* __restrict__ out, int n) {
    int tid = blockIdx.x * blockDim.x + threadIdx.x;
    if (tid < 2 * n) {
        int i = tid >> 1;
        int f = tid & 1;
        out[tid] = dis[i] * acc2[tid] + b2[f];
    }
}

extern "C" void kernel_launch(void* const* d_in, const int* in_sizes, int n_in,
                              void* d_out, int out_size, void* d_ws, size_t ws_size,
                              hipStream_t stream) {
    const float* x  = (const float*)d_in[0];
    const int*   ei = (const int*)d_in[1];
    const float* W1 = (const float*)d_in[2];
    const float* b1 = (const float*)d_in[3];
    const float* W2 = (const float*)d_in[4];
    const float* b2 = (const float*)d_in[5];
    float* out = (float*)d_out;

    const int n = in_sizes[0] / 4;    // 100000
    const int E = in_sizes[1] / 2;    // 6400000
    const int* src = ei;              // edge_index[0]
    const int* dst = ei + E;          // edge_index[1]

    // workspace layout (all f32): dis | hs1 | acc1 | hs2 | acc2  (~14.8 MB)
    float* ws   = (float*)d_ws;
    float* dis  = ws;                         // n   (deg -> rsqrt in place)
    float* hs1  = dis + n;                    // 16n
    float* acc1 = hs1 + (size_t)n * 16;       // 16n
    float* hs2  = acc1 + (size_t)n * 16;      // 2n
    float* acc2 = hs2 + (size_t)n * 2;        // 2n

    const dim3 blk(256);

    k_deg_init<<<(n + 255) / 256, blk, 0, stream>>>(dis, n);
    k_deg_edges<<<(E + 255) / 256, blk, 0, stream>>>(dst, dis, E);
    k_rsqrt_inplace<<<(n + 255) / 256, blk, 0, stream>>>(dis, n);

    const int ntile = (n + 15) / 16;
    const int tb = (ntile + 7) / 8;           // 8 waves (tiles) per 256-thread block

    k_lin1_wmma<<<tb, blk, 0, stream>>>(x, W1, dis, hs1, acc1, ntile, n);

    k_edge_scatter16<<<(E * 4 + 255) / 256, blk, 0, stream>>>(src, dst, hs1, acc1, E);

    k_lin2_wmma<<<tb, blk, 0, stream>>>(acc1, b1, W2, dis, hs2, acc2, ntile, n);

    k_edge_scatter2<<<(E + 255) / 256, blk, 0, stream>>>(src, dst, hs2, acc2, E);

    k_finish<<<(2 * n + 255) / 256, blk, 0, stream>>>(acc2, dis, b2, out, n);
}